// Model_29557964931711
// MI455X (gfx1250) — compile-verified
//
#include <hip/hip_runtime.h>
#include <hip/hip_bf16.h>

typedef __attribute__((ext_vector_type(2))) float v2f;
typedef __attribute__((ext_vector_type(8))) float v8f;

#define C 64
#define HALF 32
#define LAYERS 16
#define HID 1024
#define FEAT (18 * C) /* 1152 */
#define EPS 1e-5f

// ---------------------------------------------------------------- utilities
__global__ void k_fill_zero(float* __restrict__ p, long n) {
  long i = (long)blockIdx.x * blockDim.x + threadIdx.x;
  if (i < n) p[i] = 0.f;
}

__global__ void k_embed(const int* __restrict__ nt, const float* __restrict__ emb,
                        float* __restrict__ x, int n) {
  int i = blockIdx.x * blockDim.x + threadIdx.x;
  if (i >= n * C) return;
  int row = i >> 6, c = i & 63;
  x[i] = emb[nt[row] * C + c];
}

// gather rows `idx` of x into column slice [coloff, coloff+C) of sel [K, FEAT]
__global__ void k_gather(const float* __restrict__ x, const int* __restrict__ idx,
                         float* __restrict__ sel, int K, int coloff) {
  int i = blockIdx.x * blockDim.x + threadIdx.x;
  if (i >= K * C) return;
  int k = i >> 6, c = i & 63;
  sel[(long)k * FEAT + coloff + c] = x[(long)idx[k] * C + c];
}

// ------------------------------------------------------------ edge scatter
// aggF[dst] += x[src]; aggB[src] += x[dst]. One thread per (edge, 4-chan chunk).
__global__ void k_scatter(const float* __restrict__ x, const int* __restrict__ ei,
                          float* __restrict__ aggF, float* __restrict__ aggB, int E) {
  long t = (long)blockIdx.x * blockDim.x + threadIdx.x;
  if (t >= (long)E * 16) return;
  int e = (int)(t >> 4);
  int q = ((int)t & 15) << 2;
  int s = ei[e], d = ei[E + e];
  const float4 vs = *(const float4*)(x + (long)s * C + q);
  const float4 vd = *(const float4*)(x + (long)d * C + q);
  float* pf = aggF + (long)d * C + q;
  float* pb = aggB + (long)s * C + q;
  atomicAdd(pf + 0, vs.x); atomicAdd(pf + 1, vs.y);
  atomicAdd(pf + 2, vs.z); atomicAdd(pf + 3, vs.w);
  atomicAdd(pb + 0, vd.x); atomicAdd(pb + 1, vd.y);
  atomicAdd(pb + 2, vd.z); atomicAdd(pb + 3, vd.w);
}

// ----------------------------------------------------------------- batchnorm
// stats[0..63] = sum(relu(x)) per channel, stats[64..127] = sumsq
__global__ void k_bn_stats(const float* __restrict__ x, float* __restrict__ stats, int n) {
  int c = threadIdx.x & 63;
  int g = threadIdx.x >> 6; // 4 row-groups per block
  float s = 0.f, ss = 0.f;
  for (int row = blockIdx.x * 4 + g; row < n; row += gridDim.x * 4) {
    float v = x[(long)row * C + c];
    v = v > 0.f ? v : 0.f;
    s += v; ss += v * v;
  }
  __shared__ float shs[256], shq[256];
  shs[threadIdx.x] = s; shq[threadIdx.x] = ss;
  __syncthreads();
  if (g == 0) {
    s  = shs[c] + shs[64 + c] + shs[128 + c] + shs[192 + c];
    ss = shq[c] + shq[64 + c] + shq[128 + c] + shq[192 + c];
    atomicAdd(&stats[c], s);
    atomicAdd(&stats[64 + c], ss);
  }
}

// h = gamma*(relu(x)-mean)*rsqrt(var+eps)+beta  (biased batch var)
__global__ void k_bn_apply(const float* __restrict__ x, const float* __restrict__ stats,
                           const float* __restrict__ gamma, const float* __restrict__ beta,
                           float* __restrict__ h, int n) {
  long i = (long)blockIdx.x * blockDim.x + threadIdx.x;
  if (i >= (long)n * C) return;
  int c = (int)(i & 63);
  float inv_n = 1.f / (float)n;
  float mean = stats[c] * inv_n;
  float var  = stats[64 + c] * inv_n - mean * mean;
  float sc = gamma[c] * rsqrtf(var + EPS);
  float sh = beta[c] - mean * sc;
  float v = x[i];
  v = v > 0.f ? v : 0.f;
  h[i] = v * sc + sh;
}

// -------------------------------------------------- conv GEMM (f32 WMMA)
// out[:, dir*32 + tn*16 .. +16] tile = (dir? aggB : aggF)[M,64] @ w[dir][64,32]
// one wave per 16x16 tile; grid = (M/16, 2, 2)
__global__ void k_gemm_conv(const float* __restrict__ aggF, const float* __restrict__ aggB,
                            const float* __restrict__ w /*[2][C][HALF]*/,
                            const float* __restrict__ skip,
                            float* __restrict__ out, int M) {
  const int tm = blockIdx.x, tn = blockIdx.y, dir = blockIdx.z;
  const float* A = dir ? aggB : aggF;
  const float* B = w + dir * C * HALF;
  const int lane = threadIdx.x;
  const int l15 = lane & 15;
  const int hi  = lane >> 4; // 0: K=0,1  1: K=2,3
  int arow = tm * 16 + l15;
  if (arow >= M) arow = M - 1; // clamp loads; stores are predicated below
  const int bcol = tn * 16 + l15; // 0..31
  v8f acc = {};
#pragma unroll
  for (int k = 0; k < C; k += 4) {
    const int kk = k + hi * 2;
    v2f a, b;
    a.x = A[(long)arow * C + kk];
    a.y = A[(long)arow * C + kk + 1];
    b.x = B[kk * HALF + bcol];
    b.y = B[(kk + 1) * HALF + bcol];
    acc = __builtin_amdgcn_wmma_f32_16x16x4_f32(
        /*neg_a=*/false, a, /*neg_b=*/false, b,
        /*c_mod=*/(short)0, acc, /*reuse_a=*/false, /*reuse_b=*/false);
  }
  const int orow0 = tm * 16 + hi * 8;
  const int ocol  = dir * HALF + bcol;
#pragma unroll
  for (int r = 0; r < 8; ++r) {
    int orow = orow0 + r;
    if (orow < M) {
      float v = acc[r];
      if (skip) v += skip[(long)orow * C + ocol];
      out[(long)orow * C + ocol] = v;
    }
  }
}

// -------------------------------------------- hidden GEMM: relu(sel)@W + b, relu
// sel [K, FEAT], W [FEAT, HID]; grid = (K/16, HID/16), one wave per tile
__global__ void k_gemm_hidden(const float* __restrict__ sel, const float* __restrict__ W,
                              const float* __restrict__ bias, float* __restrict__ hid,
                              int K) {
  const int tm = blockIdx.x, tn = blockIdx.y;
  const int lane = threadIdx.x;
  const int l15 = lane & 15;
  const int hi  = lane >> 4;
  const int arow = tm * 16 + l15; // K divisible by 16
  const int bcol = tn * 16 + l15;
  v8f acc = {};
#pragma unroll 4
  for (int k = 0; k < FEAT; k += 4) {
    const int kk = k + hi * 2;
    v2f a, b;
    float a0 = sel[(long)arow * FEAT + kk];
    float a1 = sel[(long)arow * FEAT + kk + 1];
    a.x = a0 > 0.f ? a0 : 0.f;
    a.y = a1 > 0.f ? a1 : 0.f;
    b.x = W[(long)kk * HID + bcol];
    b.y = W[(long)(kk + 1) * HID + bcol];
    acc = __builtin_amdgcn_wmma_f32_16x16x4_f32(false, a, false, b, (short)0, acc,
                                                false, false);
  }
  const int orow0 = tm * 16 + hi * 8;
  const float bv = bias[bcol];
#pragma unroll
  for (int r = 0; r < 8; ++r) {
    float v = acc[r] + bv;
    v = v > 0.f ? v : 0.f; // relu(h) needed before out_w
    hid[(long)(orow0 + r) * HID + bcol] = v;
  }
}

// ------------------------------------------------------- output dot + softmax
__global__ void k_outdot(const float* __restrict__ hid, const float* __restrict__ ow,
                         const float* __restrict__ ob, float* __restrict__ o) {
  int row = blockIdx.x;
  float p = 0.f;
  for (int j = threadIdx.x; j < HID; j += 256)
    p += hid[(long)row * HID + j] * ow[j];
  __shared__ float sh[256];
  sh[threadIdx.x] = p;
  __syncthreads();
  for (int s = 128; s > 0; s >>= 1) {
    if (threadIdx.x < s) sh[threadIdx.x] += sh[threadIdx.x + s];
    __syncthreads();
  }
  if (threadIdx.x == 0) o[row] = sh[0] + ob[0];
}

__global__ void k_softmax(const float* __restrict__ o, float* __restrict__ out, int K) {
  __shared__ float sh[256];
  float m = -1e30f;
  for (int i = threadIdx.x; i < K; i += 256) m = fmaxf(m, o[i]);
  sh[threadIdx.x] = m;
  __syncthreads();
  for (int s = 128; s > 0; s >>= 1) {
    if (threadIdx.x < s) sh[threadIdx.x] = fmaxf(sh[threadIdx.x], sh[threadIdx.x + s]);
    __syncthreads();
  }
  m = sh[0];
  __syncthreads();
  float acc = 0.f;
  for (int i = threadIdx.x; i < K; i += 256) acc += __expf(o[i] - m);
  sh[threadIdx.x] = acc;
  __syncthreads();
  for (int s = 128; s > 0; s >>= 1) {
    if (threadIdx.x < s) sh[threadIdx.x] += sh[threadIdx.x + s];
    __syncthreads();
  }
  float total = sh[0];
  for (int i = threadIdx.x; i < K; i += 256) out[i] = __expf(o[i] - m) / total;
}

// ---------------------------------------------------------------- driver
extern "C" void kernel_launch(void* const* d_in, const int* in_sizes, int n_in,
                              void* d_out, int out_size, void* d_ws, size_t ws_size,
                              hipStream_t stream) {
  const int*   nt      = (const int*)d_in[0];
  const int*   ei      = (const int*)d_in[1];
  const int*   idx     = (const int*)d_in[2];
  const float* emb     = (const float*)d_in[3];
  const float* conv0_w = (const float*)d_in[4];
  const float* bng     = (const float*)d_in[5]; // [L,2,C]
  const float* bnb     = (const float*)d_in[6];
  const float* resw    = (const float*)d_in[7]; // [L,2,2,C,HALF]
  const float* hw      = (const float*)d_in[8];
  const float* hb      = (const float*)d_in[9];
  const float* ow      = (const float*)d_in[10];
  const float* ob      = (const float*)d_in[11];
  const int N = in_sizes[0];
  const int E = in_sizes[1] / 2;
  const int K = in_sizes[2];

  // workspace carve-out (256B aligned)
  char* wp = (char*)d_ws;
  auto carve = [&](size_t bytes) {
    float* p = (float*)wp;
    wp += (bytes + 255) & ~(size_t)255;
    return p;
  };
  const size_t nodeB = (size_t)N * C * sizeof(float);
  float* xA   = carve(nodeB);
  float* xB   = carve(nodeB);
  float* xT   = carve(nodeB);
  float* hbuf = carve(nodeB);
  float* aggF = carve(2 * nodeB); // aggF and aggB contiguous -> single zero-fill
  float* aggB = aggF + (size_t)N * C;
  float* stats = carve(128 * sizeof(float));
  float* sel   = carve((size_t)K * FEAT * sizeof(float));
  float* hid   = carve((size_t)K * HID * sizeof(float));
  float* ovec  = carve((size_t)K * sizeof(float));
  (void)ws_size; (void)n_in; (void)out_size;

  auto fill = [&](float* p, long n) {
    k_fill_zero<<<(unsigned)((n + 255) / 256), 256, 0, stream>>>(p, n);
  };
  dim3 gconv((N + 15) / 16, 2, 2);
  auto biconv = [&](const float* xin, const float* wmat, const float* skip, float* xout) {
    fill(aggF, 2L * N * C);
    k_scatter<<<(unsigned)(((long)E * 16 + 255) / 256), 256, 0, stream>>>(xin, ei, aggF,
                                                                          aggB, E);
    k_gemm_conv<<<gconv, 32, 0, stream>>>(aggF, aggB, wmat, skip, xout, N);
  };
  auto bn = [&](const float* xin, const float* g_, const float* b_, float* hout) {
    fill(stats, 128);
    k_bn_stats<<<512, 256, 0, stream>>>(xin, stats, N);
    k_bn_apply<<<(unsigned)(((long)N * C + 255) / 256), 256, 0, stream>>>(xin, stats, g_,
                                                                          b_, hout, N);
  };
  auto gather = [&](const float* x, int coloff) {
    k_gather<<<(unsigned)((K * C + 255) / 256), 256, 0, stream>>>(x, idx, sel, K, coloff);
  };

  // x = emb[node_types]; log0
  k_embed<<<(unsigned)((N * C + 255) / 256), 256, 0, stream>>>(nt, emb, xA, N);
  gather(xA, 0);
  // conv0; log1
  biconv(xA, conv0_w, nullptr, xB);
  gather(xB, C);

  float* xin = xB;
  float* xother = xA;
  for (int l = 0; l < LAYERS; ++l) {
    const float* g0 = bng + (size_t)(l * 2 + 0) * C;
    const float* b0 = bnb + (size_t)(l * 2 + 0) * C;
    const float* g1 = bng + (size_t)(l * 2 + 1) * C;
    const float* b1 = bnb + (size_t)(l * 2 + 1) * C;
    const float* w0 = resw + (size_t)(l * 2 + 0) * 2 * C * HALF;
    const float* w1 = resw + (size_t)(l * 2 + 1) * 2 * C * HALF;
    bn(xin, g0, b0, hbuf);
    biconv(hbuf, w0, nullptr, xT);
    bn(xT, g1, b1, hbuf);
    biconv(hbuf, w1, /*skip=*/xin, xother);
    gather(xother, (2 + l) * C);
    float* t = xin; xin = xother; xother = t;
  }

  // MLP head
  dim3 gh(K / 16, HID / 16);
  k_gemm_hidden<<<gh, 32, 0, stream>>>(sel, hw, hb, hid, K);
  k_outdot<<<K, 256, 0, stream>>>(hid, ow, ob, ovec);
  k_softmax<<<1, 256, 0, stream>>>(ovec, (float*)d_out, K);
}